// YOLOLoss_24094766530595
// MI455X (gfx1250) — compile-verified
//
#include <hip/hip_runtime.h>

// ---------------------------------------------------------------------------
// YOLO loss, MI455X (gfx1250, wave32).
// Memory-bound streaming reduction: 192.7 MB read -> 1 float (~8.3 us floor).
// - Wave-cooperative LDS staging of 120-byte AoS cells:
//     * preferred: GLOBAL_LOAD_ASYNC_TO_LDS_B128 (ASYNCcnt-tracked DMA,
//       no VGPR round-trip), 15 full-wave rounds, no masked tail.
//     * fallback: coalesced global_load_b128 + ds_store_b128.
// - Conflict-free ds_load_b64 per-lane reads (stride 30 dwords, 64 banks).
// - Cross-lane reduction on the matrix unit: v_wmma_f32_16x16x4_f32 with a
//   ones B-matrix (exact), finished by one ds_swizzle xor-16.
// ---------------------------------------------------------------------------

typedef __attribute__((ext_vector_type(2))) float v2f;
typedef __attribute__((ext_vector_type(8))) float v8f;
typedef int v4i_vs __attribute__((vector_size(16)));  // matches builtin pointee

#define TPB         256
#define NBLK        1024
#define NWAVES      (TPB / 32)
#define NCELLS      (4096 * 14 * 14)   // 802816
#define CELL_F      30                 // floats per cell
#define WAVE_F      (32 * CELL_F)      // 960 floats per tensor per wave
#define WAVE_F4     (WAVE_F / 4)       // 240 float4 per tensor per wave

#if __has_builtin(__builtin_amdgcn_global_load_async_to_lds_b128) && \
    __has_builtin(__builtin_amdgcn_s_wait_asynccnt)
#define HAVE_ASYNC_LDS 1
#else
#define HAVE_ASYNC_LDS 0
#endif

#define AS1 __attribute__((address_space(1)))
#define AS3 __attribute__((address_space(3)))

// lane (L) += lane (L^16): single DS-swizzle, xor=0x10, and=0x1f, or=0.
__device__ __forceinline__ float xor16_add(float s) {
#if __has_builtin(__builtin_amdgcn_ds_swizzle)
  s += __int_as_float(__builtin_amdgcn_ds_swizzle(__float_as_int(s), 0x401F));
#else
  s += __shfl_xor(s, 16, 32);
#endif
  return s;
}

// Exact 32-lane sum using the matrix unit.
// A(m,k): lanes 0-15 VGPR0 -> K=0, lanes 16-31 VGPR0 -> K=2 (M = lane%16).
// With a = {v, 0} and B = ones: rowsum_m = v_m + v_{m+16}. D VGPR j holds
// rows m=j (lanes 0-15) and m=8+j (lanes 16-31); summing the 8 D registers
// and adding across the xor-16 pair yields the full wave sum on every lane.
__device__ __forceinline__ float wave_sum(float v) {
#if __has_builtin(__builtin_amdgcn_wmma_f32_16x16x4_f32)
  v2f a; a[0] = v;    a[1] = 0.0f;
  v2f b; b[0] = 1.0f; b[1] = 1.0f;
  v8f c = {0.0f, 0.0f, 0.0f, 0.0f, 0.0f, 0.0f, 0.0f, 0.0f};
  v8f d = __builtin_amdgcn_wmma_f32_16x16x4_f32(
      /*neg_a=*/false, a, /*neg_b=*/false, b,
      /*c_mod=*/(short)0, c, /*reuse_a=*/false, /*reuse_b=*/false);
  float s = d[0] + d[1] + d[2] + d[3] + d[4] + d[5] + d[6] + d[7];
  return xor16_add(s);
#else
  for (int off = 16; off > 0; off >>= 1) v += __shfl_xor(v, off, 32);
  return v;
#endif
}

// Per-cell YOLO loss term (p, t are 30-float register arrays).
__device__ __forceinline__ float cell_loss(const float* __restrict__ p,
                                           const float* __restrict__ t) {
  const float CELLW = 1.0f / 14.0f;
  float coo = (t[4] > 0.0f) ? 1.0f : 0.0f;
  float noo = 1.0f - coo;

  // no-object confidence loss (conf indices 4 and 9)
  float dn0 = p[4] - t[4];
  float dn1 = p[9] - t[9];
  float noo_loss = dn0 * dn0 + dn1 * dn1;

  // target box (box 0 of target)
  float txc = t[0] * CELLW, tyc = t[1] * CELLW;
  float tx0 = txc - 0.5f * t[2], tx1 = txc + 0.5f * t[2];
  float ty0 = tyc - 0.5f * t[3], ty1 = tyc + 0.5f * t[3];
  float at  = (tx1 - tx0) * (ty1 - ty0);

  float iou[2];
#pragma unroll
  for (int b = 0; b < 2; ++b) {
    const float* q = p + b * 5;
    float pxc = q[0] * CELLW, pyc = q[1] * CELLW;
    float x0 = pxc - 0.5f * q[2], x1 = pxc + 0.5f * q[2];
    float y0 = pyc - 0.5f * q[3], y1 = pyc + 0.5f * q[3];
    float wi = fmaxf(fminf(x1, tx1) - fmaxf(x0, tx0), 0.0f);
    float hi = fmaxf(fminf(y1, ty1) - fmaxf(y0, ty0), 0.0f);
    float inter = wi * hi;
    float ap = (x1 - x0) * (y1 - y0);
    iou[b] = inter / (ap + at - inter);
  }

  // responsible box: argmax (ties -> box 0, matching jnp.argmax)
  int   r       = (iou[1] > iou[0]) ? 1 : 0;
  float max_iou = r ? iou[1] : iou[0];
  float pc_r    = r ? p[9] : p[4];
  float pc_o    = r ? p[4] : p[9];
  float dcon    = pc_r - max_iou;
  float contain     = dcon * dcon;
  float not_contain = pc_o * pc_o;

  // localization loss on responsible box (branchless selects, no reg spills)
  float prx = r ? p[5] : p[0], pry = r ? p[6] : p[1];
  float prw = r ? p[7] : p[2], prh = r ? p[8] : p[3];
  float trx = r ? t[5] : t[0], try_ = r ? t[6] : t[1];
  float trw = r ? t[7] : t[2], trh = r ? t[8] : t[3];
  float dx = prx - trx, dy = pry - try_;
  float dw = sqrtf(prw) - sqrtf(trw);
  float dh = sqrtf(prh) - sqrtf(trh);
  float loc = dx * dx + dy * dy + dw * dw + dh * dh;

  // class loss
  float cls = 0.0f;
#pragma unroll
  for (int c = 10; c < 30; ++c) {
    float d = p[c] - t[c];
    cls += d * d;
  }

  return coo * (5.0f * loc + 2.0f * contain + not_contain + cls)
       + 0.5f * noo * noo_loss;
}

__global__ __launch_bounds__(TPB) void yolo_partial_kernel(
    const float* __restrict__ pred, const float* __restrict__ targ,
    float* __restrict__ partial) {
  // Per-wave region: [pred: 960 floats][targ: 960 floats] (7680 B); 61.4 KB/block.
  __shared__ float sBuf[NWAVES][2 * WAVE_F];
  __shared__ float sRed[32];

  const int lane = threadIdx.x & 31;
  const int wave = threadIdx.x >> 5;

  if (threadIdx.x < 32) sRed[threadIdx.x] = 0.0f;
  __syncthreads();

  float acc = 0.0f;
  const int stride = NBLK * TPB;  // 262144 cells per grid sweep

  // 802816 = 3*262144 + 64*256: every wave of a block has the same trip
  // count, so EXEC stays all-1s for the WMMA reduction below.
  for (int cell0 = blockIdx.x * TPB + wave * 32; cell0 < NCELLS;
       cell0 += stride) {
    const float4* gp = (const float4*)(pred + (size_t)cell0 * CELL_F);
    // Shift targ base so "gt + idx" is valid for idx in [240, 480).
    const float4* gt = (const float4*)(targ + (size_t)cell0 * CELL_F) - WAVE_F4;
    float4* ldst = (float4*)&sBuf[wave][0];

    // 15 full-wave b128 rounds, no masked tail; base select constant-folds
    // for every unrolled round except round 7 (one cndmask there).
#pragma unroll
    for (int r = 0; r < 15; ++r) {
      int idx = r * 32 + lane;
      const float4* src = (idx < WAVE_F4) ? (gp + idx) : (gt + idx);
#if HAVE_ASYNC_LDS
      __builtin_amdgcn_global_load_async_to_lds_b128(
          (AS1 v4i_vs*)(unsigned long long)src,
          (AS3 v4i_vs*)(unsigned int)(unsigned long long)(ldst + idx),
          /*imm offset=*/0, /*cpol=*/0);
#else
      ldst[idx] = *src;  // global_load_b128 + ds_store_b128
#endif
    }
#if HAVE_ASYNC_LDS
    __builtin_amdgcn_s_wait_asynccnt(0);
#endif
    // Per-wave private regions; LDS ops from one wave execute in order,
    // so no workgroup barrier is needed in the hot loop.

    // --- each lane reads its own cell from LDS (conflict-free b64) ---
    float p[CELL_F], t[CELL_F];
    const float2* lp2 = (const float2*)&sBuf[wave][lane * CELL_F];
    const float2* lt2 = (const float2*)&sBuf[wave][WAVE_F + lane * CELL_F];
#pragma unroll
    for (int i = 0; i < 15; ++i) {
      float2 x = lp2[i]; p[2 * i] = x.x; p[2 * i + 1] = x.y;
    }
#pragma unroll
    for (int i = 0; i < 15; ++i) {
      float2 x = lt2[i]; t[2 * i] = x.x; t[2 * i + 1] = x.y;
    }

    acc += cell_loss(p, t);
  }

  // --- block reduction: wave WMMA-sum, then wave 0 sums the 8 leaders ---
  float s = wave_sum(acc);
  if (lane == 0) sRed[wave] = s;
  __syncthreads();
  if (threadIdx.x < 32) {            // wave 0 fully active -> EXEC all-1s
    float tot = wave_sum(sRed[threadIdx.x]);
    if (threadIdx.x == 0) partial[blockIdx.x] = tot;
  }
}

__global__ __launch_bounds__(TPB) void yolo_final_kernel(
    const float* __restrict__ partial, float* __restrict__ out) {
  __shared__ float sRed[32];
  const int lane = threadIdx.x & 31;
  const int wave = threadIdx.x >> 5;

  if (threadIdx.x < 32) sRed[threadIdx.x] = 0.0f;
  __syncthreads();

  float v = 0.0f;
  for (int i = threadIdx.x; i < NBLK; i += TPB) v += partial[i];  // 4 uniform iters

  float s = wave_sum(v);
  if (lane == 0) sRed[wave] = s;
  __syncthreads();
  if (threadIdx.x < 32) {
    float tot = wave_sum(sRed[threadIdx.x]);
    if (threadIdx.x == 0) out[0] = tot * (1.0f / 4096.0f);
  }
}

extern "C" void kernel_launch(void* const* d_in, const int* in_sizes, int n_in,
                              void* d_out, int out_size, void* d_ws,
                              size_t ws_size, hipStream_t stream) {
  (void)in_sizes; (void)n_in; (void)out_size; (void)ws_size;
  const float* pred = (const float*)d_in[0];
  const float* targ = (const float*)d_in[1];
  float* partial = (float*)d_ws;       // NBLK floats = 4 KB scratch
  float* out = (float*)d_out;

  yolo_partial_kernel<<<NBLK, TPB, 0, stream>>>(pred, targ, partial);
  yolo_final_kernel<<<1, TPB, 0, stream>>>(partial, out);
}